// normIt_57389353009667
// MI455X (gfx1250) — compile-verified
//
#include <hip/hip_runtime.h>
#include <stdint.h>

typedef __attribute__((ext_vector_type(2))) float v2f;
typedef __attribute__((ext_vector_type(4))) float v4f;
typedef __attribute__((ext_vector_type(8))) float v8f;
typedef __attribute__((ext_vector_type(4))) int   v4i;

#define BANDS          128
#define ROWS_PER_TILE  16
#define ROW_BYTES      (BANDS * 4)                       /* 512 */
#define LDS_ROW_BYTES  ((BANDS + 4) * 4)                 /* 528: +16B pad -> conflict-free */
#define TILE_LDS_BYTES (ROWS_PER_TILE * LDS_ROW_BYTES)   /* 8448 */
#define WAVES_PER_BLK  8
#define BLOCK_THREADS  (WAVES_PER_BLK * 32)
#define NUM_BLOCKS     1024
#define SMEM_BYTES     (WAVES_PER_BLK * 2 * TILE_LDS_BYTES) /* 135168 */

#if defined(__has_builtin)
#  if __has_builtin(__builtin_amdgcn_global_load_async_to_lds_b128) && \
      __has_builtin(__builtin_amdgcn_s_wait_asynccnt)
#    define USE_ASYNC 1
#  endif
#  if __has_builtin(__builtin_amdgcn_wmma_f32_16x16x4_f32)
#    define USE_WMMA 1
#  endif
#endif

// The async builtin takes v4i* in addrspace(1) (global) and addrspace(3) (LDS).
typedef __attribute__((address_space(1))) v4i as1_v4i;
typedef __attribute__((address_space(3))) v4i as3_v4i;

__device__ __forceinline__ as3_v4i* lds_cast(void* p) {
  // generic LDS address: low 32 bits are the LDS byte offset (ISA aperture rule)
  return (as3_v4i*)(uint32_t)(uintptr_t)p;
}
__device__ __forceinline__ as1_v4i* glob_cast(const void* p) {
  return (as1_v4i*)(uintptr_t)p;
}

// Uniform-lane broadcast via v_readlane_b32 (SGPR), not ds_bpermute.
__device__ __forceinline__ float bcast_lane(float v, int lane) {
  return __uint_as_float(__builtin_amdgcn_readlane(__float_as_uint(v), lane));
}

#ifdef USE_ASYNC
#  define WAIT_ASYNC(n) do { __builtin_amdgcn_s_wait_asynccnt(n); \
                             asm volatile("" ::: "memory"); } while (0)
#else
#  define WAIT_ASYNC(n) do { asm volatile("" ::: "memory"); } while (0)
#endif

// Copy one 16x128 f32 tile global -> LDS. One async b128 instruction per row:
// 32 lanes x 16B = 512B contiguous (fully coalesced); the async engine writes
// LDS directly (no VGPR round trip), tracked by ASYNCcnt.
__device__ __forceinline__ void stage_tile(const float* x, char* buf,
                                           long tile, int lane) {
  const char* g = (const char*)x + (size_t)tile * (ROWS_PER_TILE * ROW_BYTES)
                  + (size_t)lane * 16;
#ifdef USE_ASYNC
#pragma unroll
  for (int r = 0; r < ROWS_PER_TILE; ++r)
    __builtin_amdgcn_global_load_async_to_lds_b128(
        glob_cast(g + r * ROW_BYTES),
        lds_cast(buf + r * LDS_ROW_BYTES + lane * 16), 0, 0);
#else
#pragma unroll
  for (int r = 0; r < ROWS_PER_TILE; ++r)
    *(v4f*)(buf + r * LDS_ROW_BYTES + lane * 16) =
        *(const v4f*)(g + r * ROW_BYTES);
#endif
}

__device__ __forceinline__ void process_tile(const char* buf, float* out,
                                             long tile, int lane) {
  const int row = lane & 15;   // M index of this lane's A-matrix slot
  const int hi  = lane >> 4;   // lanes 16..31 hold K=2,3 per WMMA A layout

  // Accumulate this lane's K-slice in the exact 16x4 F32 A-matrix layout:
  // lane<16: (K0,K1) of row `row`; lane>=16: (K2,K3).  Elementwise f32 adds;
  // with B = ones the WMMA then sums the 4 K-slots per row -> exact row sum.
  v2f partial = {0.f, 0.f};
#pragma unroll
  for (int k = 0; k < BANDS; k += 4)
    partial += *(const v2f*)(buf + row * LDS_ROW_BYTES + (k + hi * 2) * 4);

#ifdef USE_WMMA
  v2f ones = {1.f, 1.f};
  v8f acc = {};
  // D = A(16x4 partial sums) x B(4x16 ones) + 0  => every D[m][n] = rowsum(m)
  acc = __builtin_amdgcn_wmma_f32_16x16x4_f32(false, partial, false, ones,
                                              (short)0, acc, false, false);
#else
  float p2 = partial.x + partial.y;
#endif

  char* o = (char*)out + (size_t)tile * (ROWS_PER_TILE * ROW_BYTES)
            + (size_t)lane * 16;
#pragma unroll
  for (int r = 0; r < ROWS_PER_TILE; ++r) {
#ifdef USE_WMMA
    // C/D layout: VGPR j holds M=j (lanes 0-15) and M=j+8 (lanes 16-31);
    // source lane is uniform -> v_readlane broadcast, no LDS traffic.
    float s = bcast_lane(acc[r & 7], (r < 8) ? 0 : 16);
#else
    float s = bcast_lane(p2, r) + bcast_lane(p2, r + 16);
#endif
    float inv = 1.0f / s;
    v4f v = *(const v4f*)(buf + r * LDS_ROW_BYTES + lane * 16);
    v *= inv;
    // streaming output: non-temporal so the 1 GiB result doesn't churn L2
    __builtin_nontemporal_store(v, (v4f*)(o + r * ROW_BYTES));
  }
}

__global__ __launch_bounds__(BLOCK_THREADS)
void normit_kernel(const float* __restrict__ x, float* __restrict__ out,
                   int n_tiles) {
  extern __shared__ char smem[];
  const int lane  = threadIdx.x & 31;
  const int waveL = threadIdx.x >> 5;
  char* myLds = smem + waveL * (2 * TILE_LDS_BYTES);

  const long n_waves = (long)gridDim.x * WAVES_PER_BLK;
  long t = (long)blockIdx.x * WAVES_PER_BLK + waveL;
  int buf = 0;
  if (t < n_tiles) stage_tile(x, myLds, t, lane);     // wave-uniform branch
  while (t < n_tiles) {                               // wave-uniform loop
    long tn = t + n_waves;
    if (tn < n_tiles) {
      stage_tile(x, myLds + (buf ^ 1) * TILE_LDS_BYTES, tn, lane);
      WAIT_ASYNC(16);   // 16 newer copies in flight; the older 16 are done
    } else {
      WAIT_ASYNC(0);
    }
    process_tile(myLds + buf * TILE_LDS_BYTES, out, t, lane);
    buf ^= 1;
    t = tn;
  }
}

extern "C" void kernel_launch(void* const* d_in, const int* in_sizes, int n_in,
                              void* d_out, int out_size, void* d_ws, size_t ws_size,
                              hipStream_t stream) {
  (void)d_ws; (void)ws_size; (void)n_in; (void)out_size;
  const float* x = (const float*)d_in[0];
  float* out = (float*)d_out;
  const int n_rows  = in_sizes[0] / BANDS;
  const int n_tiles = n_rows / ROWS_PER_TILE;
  (void)hipFuncSetAttribute(reinterpret_cast<const void*>(normit_kernel),
                            hipFuncAttributeMaxDynamicSharedMemorySize,
                            SMEM_BYTES);
  normit_kernel<<<NUM_BLOCKS, BLOCK_THREADS, SMEM_BYTES, stream>>>(x, out, n_tiles);
}